// multi_class_dice_63702954934360
// MI455X (gfx1250) — compile-verified
//
#include <hip/hip_runtime.h>
#include <hip/hip_bf16.h>

// Problem constants from the reference: B=16, C=8, H=512, W=512
#define NB 16
#define NC 8
#define NH 512
#define NW 512
#define HWSZ (NH * NW)
#define NROWS (NB * NH)   // 8192 rows, one wave32 per row

typedef float v2f __attribute__((ext_vector_type(2)));
typedef float v8f __attribute__((ext_vector_type(8)));

// Full 32-lane sum using V_WMMA_F32_16X16X4_F32 as the reducer.
// A = ones(16x4), B holds the 32 lane values in VGPR0 (VGPR1 = 0).
// D[m,n] = sum_k B[k,n]  => sum over the 16 columns of row 0 of D equals the
// total of all 64 B slots (= sum of lane values), independent of the exact
// B slot->(k,n) mapping. d[0]: lanes 0-15 = D[0,n], lanes 16-31 = D[8,n]
// (identical rows since A is all-ones), so 4 xor-shuffle adds over the 16-lane
// halves give the exact total in every lane.
__device__ __forceinline__ float wave_sum_wmma(float x) {
    v2f a; a[0] = 1.0f; a[1] = 1.0f;     // every slot of the 16x4 A matrix = 1
    v2f b; b[0] = x;    b[1] = 0.0f;
    v8f c = {};
    v8f d = __builtin_amdgcn_wmma_f32_16x16x4_f32(
        /*neg_a=*/false, a, /*neg_b=*/false, b,
        /*c_mod=*/(short)0, c, /*reuse_a=*/false, /*reuse_b=*/false);
    float v = d[0];
    v += __shfl_xor(v, 1, 32);
    v += __shfl_xor(v, 2, 32);
    v += __shfl_xor(v, 4, 32);
    v += __shfl_xor(v, 8, 32);
    return v;   // total in all lanes
}

__device__ __forceinline__ float f4e(const float4& v, int j) {
    switch (j) { case 0: return v.x; case 1: return v.y; case 2: return v.z; default: return v.w; }
}
__device__ __forceinline__ int i4e(const int4& v, int j) {
    switch (j) { case 0: return v.x; case 1: return v.y; case 2: return v.z; default: return v.w; }
}

// One wave per (b,h) row: stream 8 channels + target with float4/int4 loads,
// per-pixel softmax, accumulate per-class {intersection, denom} over W,
// WMMA-reduce across lanes, lane 0 writes the mask-weighted row loss.
__global__ __launch_bounds__(256) void dice_rows_kernel(
    const float* __restrict__ pred, const int* __restrict__ tgt,
    const float* __restrict__ mask, float* __restrict__ rowLoss)
{
    const int lane = threadIdx.x & 31;
    const int row  = (int)((blockIdx.x * blockDim.x + threadIdx.x) >> 5);  // 0..8191
    const int b = row >> 9;          // row / 512
    const int h = row & (NH - 1);    // row % 512

    const float* pr = pred + (size_t)b * NC * HWSZ + (size_t)h * NW;
    const int*   tr = tgt  + (size_t)b * HWSZ      + (size_t)h * NW;

    float accI[7], accD[7];
#pragma unroll
    for (int c = 0; c < 7; ++c) { accI[c] = 0.0f; accD[c] = 0.0f; }

#pragma unroll
    for (int i = 0; i < 4; ++i) {
        const int w = i * 128 + lane * 4;   // 4 consecutive pixels per lane

        float4 x[NC];
#pragma unroll
        for (int ch = 0; ch < NC; ++ch)
            x[ch] = *(const float4*)(pr + (size_t)ch * HWSZ + w);
        const int4 tq = *(const int4*)(tr + w);

#pragma unroll
        for (int j = 0; j < 4; ++j) {
            float pc[NC];
#pragma unroll
            for (int ch = 0; ch < NC; ++ch) pc[ch] = f4e(x[ch], j);
            const int tv = i4e(tq, j);

            float mx = pc[0];
#pragma unroll
            for (int ch = 1; ch < NC; ++ch) mx = fmaxf(mx, pc[ch]);

            float e[NC], s = 0.0f;
#pragma unroll
            for (int ch = 0; ch < NC; ++ch) { e[ch] = __expf(pc[ch] - mx); s += e[ch]; }
            const float inv = 1.0f / s;

#pragma unroll
            for (int c = 1; c < NC; ++c) {
                const float q   = e[c] * inv;                 // softmax prob of class c
                const float hit = (tv == c) ? 1.0f : 0.0f;    // y_true (y_true^2 == y_true)
                accI[c - 1] += hit * q;                       // intersection term
                accD[c - 1]  = fmaf(q, q, accD[c - 1] + hit); // y_true^2 + y_pred^2
            }
        }
    }

    // Cross-lane reductions (uniform control flow -> EXEC all ones for WMMA).
#pragma unroll
    for (int c = 0; c < 7; ++c) {
        accI[c] = wave_sum_wmma(accI[c]);
        accD[c] = wave_sum_wmma(accD[c]);
    }

    if (lane == 0) {
        float loss = 0.0f;
#pragma unroll
        for (int c = 0; c < 7; ++c) {
            const float use  = (mask[b * 7 + c] == 1.0f) ? 1.0f : 0.0f;
            const float dice = (2.0f * accI[c] + 1.0f) / (accD[c] + 1.0f);
            loss += use * (1.0f - dice);
        }
        rowLoss[row] = loss;
    }
}

// Deterministic fixed-tree reduction of the 8192 row losses -> scalar.
__global__ __launch_bounds__(256) void dice_final_kernel(
    const float* __restrict__ rowLoss, const float* __restrict__ mask,
    float* __restrict__ out)
{
    __shared__ float sbuf[8];
    const int t = (int)threadIdx.x;

    float s = 0.0f;
#pragma unroll
    for (int i = 0; i < NROWS / 256; ++i) s += rowLoss[t + i * 256];

    s = wave_sum_wmma(s);                    // per-wave total (uniform flow)
    const int lane = t & 31, w = t >> 5;
    if (lane == 0) sbuf[w] = s;
    __syncthreads();

    if (t == 0) {
        float tot = 0.0f;
#pragma unroll
        for (int i = 0; i < 8; ++i) tot += sbuf[i];
        float used = 0.0f;
        for (int i = 0; i < NB * (NC - 1); ++i)
            used += (mask[i] == 1.0f) ? 1.0f : 0.0f;
        out[0] = tot / (float)NH / used;     // mean over H, then / used_channels
    }
}

extern "C" void kernel_launch(void* const* d_in, const int* in_sizes, int n_in,
                              void* d_out, int out_size, void* d_ws, size_t ws_size,
                              hipStream_t stream) {
    const float* pred = (const float*)d_in[0];
    const int*   tgt  = (const int*)d_in[1];
    const float* mask = (const float*)d_in[2];
    float* rowLoss = (float*)d_ws;           // 8192 floats = 32 KB scratch
    float* out     = (float*)d_out;

    dice_rows_kernel<<<NROWS / 8, 256, 0, stream>>>(pred, tgt, mask, rowLoss);
    dice_final_kernel<<<1, 256, 0, stream>>>(rowLoss, mask, out);
}